// scGAT_skipcat_40106404610258
// MI455X (gfx1250) — compile-verified
//
#include <hip/hip_runtime.h>
#include <hip/hip_bf16.h>
#include <math.h>

// ---------------------------------------------------------------------------
// scGAT (3-layer GAT with skip-concat) for MI455X / gfx1250, wave32 + WMMA f32
// ---------------------------------------------------------------------------

typedef __attribute__((ext_vector_type(2))) float v2f;
typedef __attribute__((ext_vector_type(8))) float v8f;

#define NEG_SLOPE 0.2f
#define HEADS 8

// ---------------- ordered-float helpers for atomic max --------------------
__device__ __forceinline__ unsigned f2ord(float f) {
    unsigned u = __float_as_uint(f);
    return (u & 0x80000000u) ? ~u : (u | 0x80000000u);
}
__device__ __forceinline__ float ord2f(unsigned u) {
    return __uint_as_float((u & 0x80000000u) ? (u & 0x7fffffffu) : ~u);
}

// ---------------------------------------------------------------------------
// GEMM: D[M,Ncols] = A[M,K] * B[K,Ncols]  (row-major), optional accumulate.
// Each wave computes a 16x64 strip with 4x V_WMMA_F32_16X16X4_F32 per K-step.
// block = 128 threads (4 waves stacked on M); grid.y covers Ncols in 64-wide
// column groups.
// ---------------------------------------------------------------------------
template <bool ACC>
__global__ void gemm_wmma_f32_k(const float* __restrict__ A,
                                const float* __restrict__ B,
                                float* __restrict__ D,
                                int M, int K, int Ncols) {
    const int lane = threadIdx.x & 31;
    const int wave = threadIdx.x >> 5;
    const int m0 = (blockIdx.x * 4 + wave) * 16;
    if (m0 >= M) return;                      // wave-uniform branch (EXEC stays all-1s)
    const int n0 = blockIdx.y * 64;
    const int mrow = lane & 15;
    const int half = lane >> 4;

    int arow = m0 + mrow;
    if (arow >= M) arow = M - 1;              // safe read for ragged M (result discarded)
    const float* __restrict__ Arow = A + (size_t)arow * K;

    v8f acc0 = {}, acc1 = {}, acc2 = {}, acc3 = {};

    for (int k = 0; k < K; k += 4) {
        // A fragment (16x4): VGPR v holds K = k + 2*half + v for this lane's row
        v2f a;
        a.x = Arow[k + 2 * half];
        a.y = Arow[k + 2 * half + 1];
        // B fragments (4x16 each): VGPR v holds row K = k + v + 2*half, col = n
        const float* __restrict__ B0 = B + (size_t)(k + 2 * half) * Ncols + n0 + mrow;
        const float* __restrict__ B1 = B + (size_t)(k + 2 * half + 1) * Ncols + n0 + mrow;
        v2f b0, b1v, b2v, b3v;
        b0.x  = B0[0];   b0.y  = B1[0];
        b1v.x = B0[16];  b1v.y = B1[16];
        b2v.x = B0[32];  b2v.y = B1[32];
        b3v.x = B0[48];  b3v.y = B1[48];
        acc0 = __builtin_amdgcn_wmma_f32_16x16x4_f32(false, a, false, b0,  (short)0, acc0, false, false);
        acc1 = __builtin_amdgcn_wmma_f32_16x16x4_f32(false, a, false, b1v, (short)0, acc1, false, false);
        acc2 = __builtin_amdgcn_wmma_f32_16x16x4_f32(false, a, false, b2v, (short)0, acc2, false, false);
        acc3 = __builtin_amdgcn_wmma_f32_16x16x4_f32(false, a, false, b3v, (short)0, acc3, false, false);
    }

    // C/D layout: VGPR r, lane half -> row m0 + 8*half + r, col n0 + (lane&15)
#pragma unroll
    for (int r = 0; r < 8; ++r) {
        int row = m0 + half * 8 + r;
        if (row >= M) continue;
        size_t idx = (size_t)row * Ncols + n0 + mrow;
        if (ACC) {
            D[idx]      += acc0[r];
            D[idx + 16] += acc1[r];
            D[idx + 32] += acc2[r];
            D[idx + 48] += acc3[r];
        } else {
            D[idx]      = acc0[r];
            D[idx + 16] = acc1[r];
            D[idx + 32] = acc2[r];
            D[idx + 48] = acc3[r];
        }
    }
}

// ---------------------------------------------------------------------------
// Per (node, head): alpha_s/alpha_d dot products + init emax/-inf, den=0
// ---------------------------------------------------------------------------
__global__ void prep_alpha_k(const float* __restrict__ h,
                             const float* __restrict__ a_s,
                             const float* __restrict__ a_d,
                             float* __restrict__ asb, float* __restrict__ adb,
                             unsigned* __restrict__ emax, float* __restrict__ den,
                             int N, int C) {
    int tid = blockIdx.x * blockDim.x + threadIdx.x;
    if (tid >= N * HEADS) return;
    int n = tid >> 3, hh = tid & 7;
    const float* __restrict__ hp  = h + (size_t)n * HEADS * C + hh * C;
    const float* __restrict__ asw = a_s + hh * C;
    const float* __restrict__ adw = a_d + hh * C;
    float s = 0.f, d = 0.f;
    for (int c = 0; c < C; ++c) {
        float v = hp[c];
        s = fmaf(v, asw[c], s);
        d = fmaf(v, adw[c], d);
    }
    asb[tid] = s;
    adb[tid] = d;
    emax[tid] = f2ord(-__builtin_inff());
    den[tid] = 0.f;
}

__global__ void zero_k(float* __restrict__ p, size_t n) {
    size_t tid = (size_t)blockIdx.x * blockDim.x + threadIdx.x;
    if (tid < n) p[tid] = 0.f;
}

// ---------------------------------------------------------------------------
// Edge pass 1: segment max of leaky_relu(as[src]+ad[dst]) per (dst, head)
// ---------------------------------------------------------------------------
__global__ void edge_max_k(const int* __restrict__ ei, int E, int Etot,
                           const float* __restrict__ asb, const float* __restrict__ adb,
                           unsigned* __restrict__ emax) {
    int e = blockIdx.x * blockDim.x + threadIdx.x;
    if (e >= Etot) return;
    int src, dst;
    if (e < E) { src = ei[e]; dst = ei[E + e]; }
    else       { src = dst = e - E; }
    const float4* sp = (const float4*)(asb + (size_t)src * 8);
    const float4* dp = (const float4*)(adb + (size_t)dst * 8);
    float4 s0 = sp[0], s1 = sp[1], d0 = dp[0], d1 = dp[1];
    float v[8] = { s0.x + d0.x, s0.y + d0.y, s0.z + d0.z, s0.w + d0.w,
                   s1.x + d1.x, s1.y + d1.y, s1.z + d1.z, s1.w + d1.w };
#pragma unroll
    for (int hh = 0; hh < 8; ++hh) {
        float t = v[hh];
        t = (t >= 0.f) ? t : NEG_SLOPE * t;
        atomicMax(&emax[(size_t)dst * 8 + hh], f2ord(t));
    }
}

// ---------------------------------------------------------------------------
// Edge pass 2: w = exp(e - emax[dst]); stash w per edge; atomic sum into den
// ---------------------------------------------------------------------------
__global__ void edge_expsum_k(const int* __restrict__ ei, int E, int Etot,
                              const float* __restrict__ asb, const float* __restrict__ adb,
                              const unsigned* __restrict__ emax, float* __restrict__ den,
                              float* __restrict__ wbuf) {
    int e = blockIdx.x * blockDim.x + threadIdx.x;
    if (e >= Etot) return;
    int src, dst;
    if (e < E) { src = ei[e]; dst = ei[E + e]; }
    else       { src = dst = e - E; }
    const float4* sp = (const float4*)(asb + (size_t)src * 8);
    const float4* dp = (const float4*)(adb + (size_t)dst * 8);
    float4 s0 = sp[0], s1 = sp[1], d0 = dp[0], d1 = dp[1];
    float v[8] = { s0.x + d0.x, s0.y + d0.y, s0.z + d0.z, s0.w + d0.w,
                   s1.x + d1.x, s1.y + d1.y, s1.z + d1.z, s1.w + d1.w };
#pragma unroll
    for (int hh = 0; hh < 8; ++hh) {
        float t = v[hh];
        t = (t >= 0.f) ? t : NEG_SLOPE * t;
        float m = ord2f(emax[(size_t)dst * 8 + hh]);
        float w = expf(t - m);
        wbuf[(size_t)e * 8 + hh] = w;
        atomicAdd(&den[(size_t)dst * 8 + hh], w);
    }
}

// ---------------------------------------------------------------------------
// Edge pass 3: agg[dst] += h[src] * w / (den[dst] + 1e-16); 1 thread per
// (edge, head), C atomic f32 adds each.
// ---------------------------------------------------------------------------
__global__ void edge_agg_k(const int* __restrict__ ei, int E, int Etot,
                           const float* __restrict__ hbuf, const float* __restrict__ wbuf,
                           const float* __restrict__ den, float* __restrict__ agg, int C) {
    long long tid = (long long)blockIdx.x * blockDim.x + threadIdx.x;
    if (tid >= (long long)Etot * 8) return;
    int e = (int)(tid >> 3), hh = (int)(tid & 7);
    int src, dst;
    if (e < E) { src = ei[e]; dst = ei[E + e]; }
    else       { src = dst = e - E; }
    float alpha = wbuf[tid] / (den[(size_t)dst * 8 + hh] + 1e-16f);
    const float* __restrict__ hp = hbuf + (size_t)src * HEADS * C + hh * C;
    float* __restrict__ op = agg + (size_t)dst * HEADS * C + hh * C;
    for (int c = 0; c < C; ++c) {
        atomicAdd(&op[c], hp[c] * alpha);
    }
}

// ---------------------------------------------------------------------------
// elu(x + bias)
// ---------------------------------------------------------------------------
__global__ void elu_bias_k(const float* __restrict__ agg, const float* __restrict__ b,
                           float* __restrict__ out, size_t n, int hdim) {
    size_t tid = (size_t)blockIdx.x * blockDim.x + threadIdx.x;
    if (tid >= n) return;
    float v = agg[tid] + b[tid % hdim];
    out[tid] = (v > 0.f) ? v : expm1f(v);
}

// ---------------------------------------------------------------------------
// Final: head-mean + b3 + log_softmax over 16 classes
// ---------------------------------------------------------------------------
__global__ void final_k(const float* __restrict__ agg, const float* __restrict__ b3,
                        float* __restrict__ out, int N) {
    int n = blockIdx.x * blockDim.x + threadIdx.x;
    if (n >= N) return;
    const float* __restrict__ ap = agg + (size_t)n * 128;  // [8 heads][16]
    float t[16];
#pragma unroll
    for (int c = 0; c < 16; ++c) {
        float s = 0.f;
#pragma unroll
        for (int hh = 0; hh < 8; ++hh) s += ap[hh * 16 + c];
        t[c] = s * 0.125f + b3[c];
    }
    float m = t[0];
#pragma unroll
    for (int c = 1; c < 16; ++c) m = fmaxf(m, t[c]);
    float lse = 0.f;
#pragma unroll
    for (int c = 0; c < 16; ++c) lse += expf(t[c] - m);
    lse = logf(lse);
    float* __restrict__ op = out + (size_t)n * 16;
#pragma unroll
    for (int c = 0; c < 16; ++c) op[c] = t[c] - m - lse;
}

// ---------------------------------------------------------------------------
// Host-side launcher
// ---------------------------------------------------------------------------
extern "C" void kernel_launch(void* const* d_in, const int* in_sizes, int n_in,
                              void* d_out, int out_size, void* d_ws, size_t ws_size,
                              hipStream_t stream) {
    const float* x   = (const float*)d_in[0];
    const int*   ei  = (const int*)d_in[1];
    const float* W1  = (const float*)d_in[2];
    const float* a1s = (const float*)d_in[3];
    const float* a1d = (const float*)d_in[4];
    const float* b1  = (const float*)d_in[5];
    const float* W2  = (const float*)d_in[6];
    const float* a2s = (const float*)d_in[7];
    const float* a2d = (const float*)d_in[8];
    const float* b2  = (const float*)d_in[9];
    const float* W3  = (const float*)d_in[10];
    const float* a3s = (const float*)d_in[11];
    const float* a3d = (const float*)d_in[12];
    const float* b3  = (const float*)d_in[13];
    float* out = (float*)d_out;

    const int N = in_sizes[0] / 256;   // 100000
    const int E = in_sizes[1] / 2;     // 1600000
    const int Etot = E + N;

    // workspace layout (floats)
    float* ws = (float*)d_ws;
    float*    hbuf   = ws;                               // [N,128]
    float*    aggbuf = hbuf   + (size_t)N * 128;         // [N,128]
    float*    actbuf = aggbuf + (size_t)N * 128;         // [N,64]
    float*    asb    = actbuf + (size_t)N * 64;          // [N,8]
    float*    adb    = asb    + (size_t)N * 8;           // [N,8]
    unsigned* emaxb  = (unsigned*)(adb + (size_t)N * 8); // [N,8]
    float*    denb   = (float*)(emaxb + (size_t)N * 8);  // [N,8]
    float*    wbuf   = denb   + (size_t)N * 8;           // [Etot,8]

    const int TB = 256;
    const int mtiles = (N + 15) / 16;
    dim3 gemm_block(128, 1, 1);
    dim3 gemm_grid1((mtiles + 3) / 4, 1, 1);   // Ncols=64
    dim3 gemm_grid3((mtiles + 3) / 4, 2, 1);   // Ncols=128

    int gNH   = (N * HEADS + TB - 1) / TB;
    int gE    = (Etot + TB - 1) / TB;
    long long eh = (long long)Etot * 8;
    int gEH   = (int)((eh + TB - 1) / TB);
    size_t n64  = (size_t)N * 64;
    size_t n128 = (size_t)N * 128;
    int gN64  = (int)((n64 + TB - 1) / TB);
    int gN128 = (int)((n128 + TB - 1) / TB);
    int gN    = (N + TB - 1) / TB;

    // ---------------- layer 1: DIN=256 -> 8x8, concat ----------------
    gemm_wmma_f32_k<false><<<gemm_grid1, gemm_block, 0, stream>>>(x, W1, hbuf, N, 256, 64);
    prep_alpha_k<<<gNH, TB, 0, stream>>>(hbuf, a1s, a1d, asb, adb, emaxb, denb, N, 8);
    zero_k<<<gN64, TB, 0, stream>>>(aggbuf, n64);
    edge_max_k<<<gE, TB, 0, stream>>>(ei, E, Etot, asb, adb, emaxb);
    edge_expsum_k<<<gE, TB, 0, stream>>>(ei, E, Etot, asb, adb, emaxb, denb, wbuf);
    edge_agg_k<<<gEH, TB, 0, stream>>>(ei, E, Etot, hbuf, wbuf, denb, aggbuf, 8);
    elu_bias_k<<<gN64, TB, 0, stream>>>(aggbuf, b1, actbuf, n64, 64);

    // ---------------- layer 2: [x, h1] (320) -> 8x8, concat ----------
    // concat handled as split GEMM: x @ W2[0:256,:] then h1 @ W2[256:320,:] (+=)
    gemm_wmma_f32_k<false><<<gemm_grid1, gemm_block, 0, stream>>>(x, W2, hbuf, N, 256, 64);
    gemm_wmma_f32_k<true><<<gemm_grid1, gemm_block, 0, stream>>>(actbuf, W2 + (size_t)256 * 64, hbuf, N, 64, 64);
    prep_alpha_k<<<gNH, TB, 0, stream>>>(hbuf, a2s, a2d, asb, adb, emaxb, denb, N, 8);
    zero_k<<<gN64, TB, 0, stream>>>(aggbuf, n64);
    edge_max_k<<<gE, TB, 0, stream>>>(ei, E, Etot, asb, adb, emaxb);
    edge_expsum_k<<<gE, TB, 0, stream>>>(ei, E, Etot, asb, adb, emaxb, denb, wbuf);
    edge_agg_k<<<gEH, TB, 0, stream>>>(ei, E, Etot, hbuf, wbuf, denb, aggbuf, 8);
    elu_bias_k<<<gN64, TB, 0, stream>>>(aggbuf, b2, actbuf, n64, 64);

    // ---------------- layer 3: 64 -> 8x16, head-mean + log_softmax ---
    gemm_wmma_f32_k<false><<<gemm_grid3, gemm_block, 0, stream>>>(actbuf, W3, hbuf, N, 64, 128);
    prep_alpha_k<<<gNH, TB, 0, stream>>>(hbuf, a3s, a3d, asb, adb, emaxb, denb, N, 16);
    zero_k<<<gN128, TB, 0, stream>>>(aggbuf, n128);
    edge_max_k<<<gE, TB, 0, stream>>>(ei, E, Etot, asb, adb, emaxb);
    edge_expsum_k<<<gE, TB, 0, stream>>>(ei, E, Etot, asb, adb, emaxb, denb, wbuf);
    edge_agg_k<<<gEH, TB, 0, stream>>>(ei, E, Etot, hbuf, wbuf, denb, aggbuf, 16);
    final_k<<<gN, TB, 0, stream>>>(aggbuf, b3, out, N);
}